// HyperCoCoFusion_25933012533388
// MI455X (gfx1250) — compile-verified
//
#include <hip/hip_runtime.h>
#include <math.h>

#define Nn    4096
#define Dd    256
#define OUTD  256
#define KK    4
#define TOPKV 16
#define EPSV  1e-9f

typedef float v2f __attribute__((ext_vector_type(2)));
typedef float v8f __attribute__((ext_vector_type(8)));

#define WMMA_F32(A_, B_, C_) \
    __builtin_amdgcn_wmma_f32_16x16x4_f32(false, (A_), false, (B_), (short)0, (C_), false, false)

// ---------------------------------------------------------------------------
// fp32 WMMA GEMM:  C = scale * op(A,B) [+ C] [+ bias]
//   A: M x Kd row-major (lda)
//   B: transB ? (Ncols x Kd, row-major, contracted over Kd)  (NT)
//            : (Kd x Ncols, row-major)                       (NN)
// Each wave computes a 32x32 output block as 2x2 WMMA tiles (4 independent
// accumulators -> 4-way matrix-pipe ILP, 1 float2 load per V_WMMA_F32_16X16X4).
// 8 waves per 256-thread block. M, Ncols multiples of 32; Kd multiple of 16.
// ---------------------------------------------------------------------------
__global__ __launch_bounds__(256) void hcf_gemm_f32(
    const float* __restrict__ A, int lda,
    const float* __restrict__ B, int ldb,
    float* __restrict__ C, int ldc,
    int M, int Ncols, int Kd, int transB,
    float cscale, const float* __restrict__ sptr, int sidx,
    int accflag, const float* __restrict__ bias)
{
    const int wave = threadIdx.x >> 5;
    const int lane = threadIdx.x & 31;
    const int tilesN = Ncols >> 5;
    const int tiles  = (M >> 5) * tilesN;
    const int tile   = blockIdx.x * 8 + wave;
    if (tile >= tiles) return;                 // wave-uniform: EXEC stays all-ones
    const int tm = tile / tilesN, tn = tile % tilesN;
    const int r0 = tm << 5, c0 = tn << 5;
    const int ni = lane & 15;                  // M-row for A frags, N-col for B/C frags
    const int hi = lane >> 4;                  // K-half selector

    v8f acc00 = {}, acc01 = {}, acc10 = {}, acc11 = {};
    const float* Arow0 = A + (size_t)(r0 + ni) * lda + 2 * hi;
    const float* Arow1 = Arow0 + (size_t)16 * lda;
    if (transB) {
        const float* Brow0 = B + (size_t)(c0 + ni) * ldb + 2 * hi;
        const float* Brow1 = Brow0 + (size_t)16 * ldb;
#pragma unroll 4
        for (int kb = 0; kb < Kd; kb += 4) {
            v2f a0 = *(const v2f*)(Arow0 + kb);
            v2f a1 = *(const v2f*)(Arow1 + kb);
            v2f b0 = *(const v2f*)(Brow0 + kb);
            v2f b1 = *(const v2f*)(Brow1 + kb);
            acc00 = WMMA_F32(a0, b0, acc00);
            acc01 = WMMA_F32(a0, b1, acc01);
            acc10 = WMMA_F32(a1, b0, acc10);
            acc11 = WMMA_F32(a1, b1, acc11);
        }
    } else {
        const float* Bc = B + (size_t)(2 * hi) * ldb + (c0 + ni);
#pragma unroll 4
        for (int kb = 0; kb < Kd; kb += 4) {
            v2f a0 = *(const v2f*)(Arow0 + kb);
            v2f a1 = *(const v2f*)(Arow1 + kb);
            const float* bp = Bc + (size_t)kb * ldb;
            v2f b0; b0.x = bp[0];  b0.y = bp[ldb];
            v2f b1; b1.x = bp[16]; b1.y = bp[ldb + 16];
            acc00 = WMMA_F32(a0, b0, acc00);
            acc01 = WMMA_F32(a0, b1, acc01);
            acc10 = WMMA_F32(a1, b0, acc10);
            acc11 = WMMA_F32(a1, b1, acc11);
        }
    }

    const float s = cscale * (sptr ? sptr[sidx] : 1.0f);
    v8f accs[4] = { acc00, acc01, acc10, acc11 };
#pragma unroll
    for (int q = 0; q < 4; ++q) {
        const int dm = q >> 1, dn = q & 1;
        const int col = c0 + 16 * dn + ni;
        const float bcol = bias ? bias[col] : 0.0f;
#pragma unroll
        for (int r = 0; r < 8; ++r) {
            const int row = r0 + 16 * dm + r + 8 * hi;
            float v = accs[q][r] * s + bcol;
            float* cp = C + (size_t)row * ldc + col;
            if (accflag) v += *cp;
            *cp = v;
        }
    }
}

// ---------------------------------------------------------------------------
// misc elementwise / reduction kernels
// ---------------------------------------------------------------------------
__global__ void hcf_fill0(float* p, int n) {
    for (int i = blockIdx.x * 256 + threadIdx.x; i < n; i += gridDim.x * 256) p[i] = 0.0f;
}

__global__ void hcf_softmax_alpha(const float* __restrict__ alpha, float* __restrict__ wsoft) {
    if (threadIdx.x == 0 && blockIdx.x == 0) {
        float m = alpha[0];
        for (int k = 1; k < KK; ++k) m = fmaxf(m, alpha[k]);
        float e[KK], s = 0.0f;
        for (int k = 0; k < KK; ++k) { e[k] = expf(alpha[k] - m); s += e[k]; }
        for (int k = 0; k < KK; ++k) wsoft[k] = e[k] / s;
    }
}

__global__ __launch_bounds__(256) void hcf_softmax_rows(float* __restrict__ H, int ncols) {
    __shared__ float red[256];
    const int row = blockIdx.x, t = threadIdx.x;
    float* Hr = H + (size_t)row * ncols;
    float mx = -INFINITY;
    for (int j = t; j < ncols; j += 256) mx = fmaxf(mx, Hr[j]);
    red[t] = mx; __syncthreads();
    for (int s = 128; s; s >>= 1) { if (t < s) red[t] = fmaxf(red[t], red[t + s]); __syncthreads(); }
    const float m = red[0]; __syncthreads();
    float sum = 0.0f;
    for (int j = t; j < ncols; j += 256) { float e = expf(Hr[j] - m); Hr[j] = e; sum += e; }
    red[t] = sum; __syncthreads();
    for (int s = 128; s; s >>= 1) { if (t < s) red[t] += red[t + s]; __syncthreads(); }
    const float inv = 1.0f / red[0];
    for (int j = t; j < ncols; j += 256) Hr[j] *= inv;
}

// per-row top-16 with renormalization (256 threads, 16 elems each, LDS-resident)
__global__ __launch_bounds__(256) void hcf_topk_rows(const float* __restrict__ H,
                                                     int* __restrict__ tki,
                                                     float* __restrict__ tkv) {
    __shared__ float lv[256 * 16];
    __shared__ float rv[256];
    __shared__ int   rc[256];
    __shared__ float ov[TOPKV];
    __shared__ int   oi[TOPKV];
    __shared__ float ssum;
    const int row = blockIdx.x, t = threadIdx.x;
    const float* Hr = H + (size_t)row * Nn;
    for (int j = 0; j < 16; ++j) lv[t * 16 + j] = Hr[t + 256 * j];
    __syncthreads();
    for (int r = 0; r < TOPKV; ++r) {
        float best = -INFINITY; int bc = 0x7fffffff;
        for (int j = 0; j < 16; ++j) {
            float x = lv[t * 16 + j]; int c = t + 256 * j;
            if (x > best || (x == best && c < bc)) { best = x; bc = c; }
        }
        rv[t] = best; rc[t] = bc; __syncthreads();
        for (int s = 128; s; s >>= 1) {
            if (t < s) {
                if (rv[t + s] > rv[t] || (rv[t + s] == rv[t] && rc[t + s] < rc[t])) {
                    rv[t] = rv[t + s]; rc[t] = rc[t + s];
                }
            }
            __syncthreads();
        }
        if (t == 0) {
            int c = rc[0];
            ov[r] = rv[0]; oi[r] = c;
            lv[(c & 255) * 16 + (c >> 8)] = -INFINITY;  // claim
        }
        __syncthreads();
    }
    if (t == 0) {
        float s = 0.0f;
        for (int r = 0; r < TOPKV; ++r) s += ov[r];
        ssum = s + EPSV;
    }
    __syncthreads();
    if (t < TOPKV) {
        tkv[(size_t)row * TOPKV + t] = ov[t] / ssum;
        tki[(size_t)row * TOPKV + t] = oi[t];
    }
}

// one block per hyperedge: S += outer(v,v)/De ; Dv += v
__global__ __launch_bounds__(256) void hcf_scatter_S(const int* __restrict__ tki,
                                                     const float* __restrict__ tkv,
                                                     float* __restrict__ S,
                                                     float* __restrict__ Dv) {
    __shared__ float v[TOPKV];
    __shared__ int   ix[TOPKV];
    __shared__ float de;
    const int e = blockIdx.x, t = threadIdx.x;
    if (t < TOPKV) { v[t] = tkv[(size_t)e * TOPKV + t]; ix[t] = tki[(size_t)e * TOPKV + t]; }
    __syncthreads();
    if (t == 0) {
        float s = 0.0f;
        for (int j = 0; j < TOPKV; ++j) s += v[j];
        de = s + EPSV;          // De = B.sum(0) + EPS
    }
    __syncthreads();
    const int j1 = t >> 4, j2 = t & 15;
    atomicAdd(&S[(size_t)ix[j1] * Nn + ix[j2]], v[j1] * v[j2] / de);
    if (t < TOPKV) atomicAdd(&Dv[ix[t]], v[t]);   // W == 1
}

__global__ void hcf_dvis(const float* __restrict__ Dv, float* __restrict__ Dvis, int n) {
    int i = blockIdx.x * 256 + threadIdx.x;
    if (i < n) Dvis[i] = rsqrtf(Dv[i] + EPSV);    // Dv_total = acc + EPS
}

__global__ void hcf_vinit(const float* __restrict__ v0, float* __restrict__ v,
                          float* __restrict__ scal, int n) {
    int i = blockIdx.x * 256 + threadIdx.x;
    if (i < n) v[i] = v0[i];
    if (i == 0) scal[0] = 0.0f;
}

__global__ void hcf_uprep(const float* __restrict__ v, const float* __restrict__ Dvis,
                          float* __restrict__ u, float* __restrict__ scal, int n) {
    int i = blockIdx.x * 256 + threadIdx.x;
    if (i < n) u[i] = Dvis[i] * v[i];
    if (i == 0) scal[0] = 0.0f;
}

// w = v - Dvis .* (S @ u)      (one block per row)
__global__ __launch_bounds__(256) void hcf_matvec_L(const float* __restrict__ S,
                                                    const float* __restrict__ u,
                                                    const float* __restrict__ v,
                                                    const float* __restrict__ Dvis,
                                                    float* __restrict__ w) {
    __shared__ float red[256];
    const int row = blockIdx.x, t = threadIdx.x;
    const float* Sr = S + (size_t)row * Nn;
    float acc = 0.0f;
    for (int j = t; j < Nn; j += 256) acc += Sr[j] * u[j];
    red[t] = acc; __syncthreads();
    for (int s = 128; s; s >>= 1) { if (t < s) red[t] += red[t + s]; __syncthreads(); }
    if (t == 0) w[row] = v[row] - Dvis[row] * red[0];
}

__global__ void hcf_reduce_sq(const float* __restrict__ x, int n, float* __restrict__ out) {
    __shared__ float red[256];
    const int t = threadIdx.x;
    float s = 0.0f;
    for (int j = blockIdx.x * 256 + t; j < n; j += gridDim.x * 256) s += x[j] * x[j];
    red[t] = s; __syncthreads();
    for (int st = 128; st; st >>= 1) { if (t < st) red[t] += red[t + st]; __syncthreads(); }
    if (t == 0) atomicAdd(out, red[0]);
}

__global__ void hcf_reduce_dot(const float* __restrict__ x, const float* __restrict__ y,
                               int n, float* __restrict__ out) {
    __shared__ float red[256];
    const int t = threadIdx.x;
    float s = 0.0f;
    for (int j = blockIdx.x * 256 + t; j < n; j += gridDim.x * 256) s += x[j] * y[j];
    red[t] = s; __syncthreads();
    for (int st = 128; st; st >>= 1) { if (t < st) red[t] += red[t + st]; __syncthreads(); }
    if (t == 0) atomicAdd(out, red[0]);
}

__global__ void hcf_scale_vec(float* __restrict__ v, const float* __restrict__ w,
                              const float* __restrict__ scal, int n) {
    int i = blockIdx.x * 256 + threadIdx.x;
    if (i < n) v[i] = w[i] / (sqrtf(scal[0]) + EPSV);
}

__global__ void hcf_lam_fin(float* scal) {
    if (threadIdx.x == 0 && blockIdx.x == 0) scal[1] = fmaxf(scal[0], 1e-3f);
}

// U = Dvis[row] .* X
__global__ void hcf_ew1(const float* __restrict__ X, const float* __restrict__ Dvis,
                        float* __restrict__ U) {
    for (int i = blockIdx.x * 256 + threadIdx.x; i < Nn * Dd; i += gridDim.x * 256)
        U[i] = Dvis[i >> 8] * X[i];
}

// T1 = (2/lam)*(X - Dvis.*P) - X ; U = Dvis.*T1
__global__ void hcf_ew2(const float* __restrict__ X, const float* __restrict__ P,
                        const float* __restrict__ Dvis, const float* __restrict__ scal,
                        float* __restrict__ T1, float* __restrict__ U) {
    const float c2 = 2.0f / scal[1];
    for (int i = blockIdx.x * 256 + threadIdx.x; i < Nn * Dd; i += gridDim.x * 256) {
        const int row = i >> 8;
        const float t1 = c2 * (X[i] - Dvis[row] * P[i]) - X[i];
        T1[i] = t1;
        U[i]  = Dvis[row] * t1;
    }
}

// T2 = 2*((2/lam)*(T1 - Dvis.*P2) - T1) - X ; out = X*th0 + T1*th1 + T2*th2
// Zmix = sigmoid(rho)*elu(out) + (1-sigmoid(rho))*Zkern
__global__ void hcf_ew3(const float* __restrict__ X, const float* __restrict__ T1,
                        const float* __restrict__ P2, const float* __restrict__ Dvis,
                        const float* __restrict__ scal, const float* __restrict__ theta,
                        const float* __restrict__ rho_raw, const float* __restrict__ Zkern,
                        float* __restrict__ Zmix) {
    const float c2  = 2.0f / scal[1];
    const float rho = 1.0f / (1.0f + expf(-rho_raw[0]));
    for (int i = blockIdx.x * 256 + threadIdx.x; i < Nn * Dd; i += gridDim.x * 256) {
        const int row = i >> 8, col = i & 255;
        const float t1  = T1[i];
        const float lt1 = c2 * (t1 - Dvis[row] * P2[i]) - t1;
        const float t2  = 2.0f * lt1 - X[i];
        const float o   = X[i] * theta[col] + t1 * theta[Dd + col] + t2 * theta[2 * Dd + col];
        const float zs  = (o > 0.0f) ? o : (expf(o) - 1.0f);   // elu
        Zmix[i] = rho * zs + (1.0f - rho) * Zkern[i];
    }
}

// ---------------------------------------------------------------------------
static inline void launch_gemm(hipStream_t s, const float* A, int lda, const float* B, int ldb,
                               float* C, int ldc, int M, int Ncols, int Kd, int transB,
                               float cs, const float* sp, int si, int acc, const float* bias) {
    const int tiles  = (M / 32) * (Ncols / 32);   // 32x32 per wave
    const int blocks = (tiles + 7) / 8;
    hcf_gemm_f32<<<blocks, 256, 0, s>>>(A, lda, B, ldb, C, ldc, M, Ncols, Kd, transB,
                                        cs, sp, si, acc, bias);
}

extern "C" void kernel_launch(void* const* d_in, const int* in_sizes, int n_in,
                              void* d_out, int out_size, void* d_ws, size_t ws_size,
                              hipStream_t stream) {
    (void)in_sizes; (void)n_in; (void)out_size; (void)ws_size;
    const float* X       = (const float*)d_in[0];   // (N,D)
    const float* Lk      = (const float*)d_in[1];   // (K,D,D)
    const float* alpha   = (const float*)d_in[2];   // (K,)
    const float* theta   = (const float*)d_in[3];   // (3,D)
    const float* rho_raw = (const float*)d_in[4];   // ()
    const float* proj_w  = (const float*)d_in[5];   // (OUT,D)
    const float* proj_b  = (const float*)d_in[6];   // (OUT,)
    const float* v0      = (const float*)d_in[7];   // (N,1)
    float* out = (float*)d_out;                     // (N,OUT)
    float* W = (float*)d_ws;

    // workspace layout (floats)
    const size_t oA    = 0;                               // N*N : A/H, later reused as S
    const size_t oYb   = oA    + (size_t)Nn * Nn;         // N*D
    const size_t oZk   = oYb   + (size_t)Nn * Dd;         // N*D
    const size_t oT1   = oZk   + (size_t)Nn * Dd;         // N*D
    const size_t oP    = oT1   + (size_t)Nn * Dd;         // N*D
    const size_t oU    = oP    + (size_t)Nn * Dd;         // N*D
    const size_t oZm   = oU    + (size_t)Nn * Dd;         // N*D
    const size_t oTKv  = oZm   + (size_t)Nn * Dd;         // K*N*16
    const size_t oTKi  = oTKv  + (size_t)KK * Nn * TOPKV; // K*N*16 (ints)
    const size_t oDv   = oTKi  + (size_t)KK * Nn * TOPKV; // N
    const size_t oDvis = oDv   + Nn;                      // N
    const size_t oV    = oDvis + Nn;                      // N
    const size_t oWv   = oV    + Nn;                      // N
    const size_t oUv   = oWv   + Nn;                      // N
    const size_t oWs   = oUv   + Nn;                      // 64 (alpha softmax)
    const size_t oScal = oWs   + 64;                      // 64 (scalars)

    float* Abuf  = W + oA;      // also S
    float* Ybuf  = W + oYb;
    float* Zkern = W + oZk;
    float* T1    = W + oT1;
    float* P     = W + oP;
    float* U     = W + oU;
    float* Zmix  = W + oZm;
    float* TKv   = W + oTKv;
    int*   TKi   = (int*)(W + oTKi);
    float* Dv    = W + oDv;
    float* Dvis  = W + oDvis;
    float* v     = W + oV;
    float* wv    = W + oWv;
    float* uv    = W + oUv;
    float* wsoft = W + oWs;
    float* scal  = W + oScal;

    // --- stage 0: init ---
    hcf_fill0<<<4096, 256, 0, stream>>>(Zkern, Nn * Dd);
    hcf_softmax_alpha<<<1, 32, 0, stream>>>(alpha, wsoft);

    // --- stage 1: per-kernel attention stack ---
    for (int k = 0; k < KK; ++k) {
        // Y = (X @ L_k) / sqrt(D)      (NN, M=N, N=D, K=D)
        launch_gemm(stream, X, Dd, Lk + (size_t)k * Dd * Dd, Dd, Ybuf, Dd,
                    Nn, Dd, Dd, /*transB=*/0, 0.0625f, nullptr, 0, 0, nullptr);
        // A = Y @ X^T                  (NT, M=N, N=N, K=D)
        launch_gemm(stream, Ybuf, Dd, X, Dd, Abuf, Nn,
                    Nn, Nn, Dd, /*transB=*/1, 1.0f, nullptr, 0, 0, nullptr);
        // row softmax in place -> H_k
        hcf_softmax_rows<<<Nn, 256, 0, stream>>>(Abuf, Nn);
        // per-row top-16 (renormalized)
        hcf_topk_rows<<<Nn, 256, 0, stream>>>(Abuf, TKi + (size_t)k * Nn * TOPKV,
                                              TKv + (size_t)k * Nn * TOPKV);
        // Zkern += softmax(alpha)[k] * H_k @ X   (NN, M=N, N=D, K=N)
        launch_gemm(stream, Abuf, Nn, X, Dd, Zkern, Dd,
                    Nn, Dd, Nn, /*transB=*/0, 1.0f, wsoft, k, /*acc=*/1, nullptr);
    }

    // --- stage 2: hypergraph operator S (reuse Abuf) and vertex degrees ---
    hcf_fill0<<<65536, 256, 0, stream>>>(Abuf, Nn * Nn);
    hcf_fill0<<<16, 256, 0, stream>>>(Dv, Nn);
    hcf_scatter_S<<<KK * Nn, 256, 0, stream>>>(TKi, TKv, Abuf, Dv);
    hcf_dvis<<<16, 256, 0, stream>>>(Dv, Dvis, Nn);

    // --- stage 3: power iteration for lambda_max ---
    hcf_vinit<<<16, 256, 0, stream>>>(v0, v, scal, Nn);
    hcf_reduce_sq<<<16, 256, 0, stream>>>(v, Nn, scal);
    hcf_scale_vec<<<16, 256, 0, stream>>>(v, v, scal, Nn);   // v /= (||v0||+eps)
    for (int it = 0; it < 5; ++it) {
        hcf_uprep<<<16, 256, 0, stream>>>(v, Dvis, uv, scal, Nn);
        hcf_matvec_L<<<Nn, 256, 0, stream>>>(Abuf, uv, v, Dvis, wv);   // wv = L v
        hcf_reduce_sq<<<16, 256, 0, stream>>>(wv, Nn, scal);
        hcf_scale_vec<<<16, 256, 0, stream>>>(v, wv, scal, Nn);
    }
    hcf_uprep<<<16, 256, 0, stream>>>(v, Dvis, uv, scal, Nn);
    hcf_matvec_L<<<Nn, 256, 0, stream>>>(Abuf, uv, v, Dvis, wv);
    hcf_reduce_dot<<<16, 256, 0, stream>>>(v, wv, Nn, scal);           // lam = v^T L v
    hcf_lam_fin<<<1, 32, 0, stream>>>(scal);                           // scal[1]=max(lam,1e-3)

    // --- stage 4: Chebyshev filter via two S@U WMMA GEMMs ---
    hcf_ew1<<<4096, 256, 0, stream>>>(X, Dvis, U);                     // U = Dvis.*X
    launch_gemm(stream, Abuf, Nn, U, Dd, P, Dd, Nn, Dd, Nn, 0, 1.0f, nullptr, 0, 0, nullptr);
    hcf_ew2<<<4096, 256, 0, stream>>>(X, P, Dvis, scal, T1, U);        // T1, U=Dvis.*T1
    launch_gemm(stream, Abuf, Nn, U, Dd, P, Dd, Nn, Dd, Nn, 0, 1.0f, nullptr, 0, 0, nullptr);
    hcf_ew3<<<4096, 256, 0, stream>>>(X, T1, P, Dvis, scal, theta, rho_raw, Zkern, Zmix);

    // --- stage 5: final projection  out = Zmix @ proj_w^T + proj_b  (NT) ---
    launch_gemm(stream, Zmix, Dd, proj_w, Dd, out, OUTD,
                Nn, OUTD, Dd, /*transB=*/1, 1.0f, nullptr, 0, 0, proj_b);
}